// ContinuousThoughtMachineSIMPLE_20607253086661
// MI455X (gfx1250) — compile-verified
//
#include <hip/hip_runtime.h>
#include <math.h>

// ---------------------------------------------------------------------------
// Continuous Thought Machine for MI455X (gfx1250), wave32, WMMA f32 16x16x4.
//   B=128, DIN=512, DM=2048, MEM=25, HID=32, NS=256, OUT=1000, ITERS=24
//   SYNC = 256*257/2 = 32896
// A matrices are stored k-pair interleaved:  A[(k>>1)*256 + m*2 + (k&1)]
// so a WMMA A-fragment (m, k..k+1) is one aligned global_load_b64.
// B tiles are staged global->LDS with CDNA5 async loads (ASYNCcnt).
// ---------------------------------------------------------------------------

#define BB      128
#define DIN     512
#define DM      2048
#define MEMN    25
#define HIDN    32
#define NSY     256
#define OUTD    1000
#define ITERS   24
#define SYNCN   32896
#define KS_OUT  16   // K-splits for the out GEMM
#define KS_SYN  2    // K-splits for the synapse GEMM

typedef __attribute__((ext_vector_type(2))) float v2f;
typedef __attribute__((ext_vector_type(8))) float v8f;

#if defined(__AMDGCN__) && __has_builtin(__builtin_amdgcn_wmma_f32_16x16x4_f32)
#define HAVE_WMMA_F32 1
#else
#define HAVE_WMMA_F32 0
#endif

// CDNA5 async global->LDS staging (ASYNCcnt-tracked), guarded.
#if defined(__AMDGCN__) && \
    __has_builtin(__builtin_amdgcn_global_load_async_to_lds_b128)
#define ASYNC_LDS 1
typedef int v4i __attribute__((vector_size(16)));
typedef __attribute__((address_space(1))) v4i* as1_v4i_p;
typedef __attribute__((address_space(3))) v4i* as3_v4i_p;
__device__ __forceinline__ void async_stage16(const float* g, float* l) {
  __builtin_amdgcn_global_load_async_to_lds_b128((as1_v4i_p)g, (as3_v4i_p)l, 0,
                                                 0);
}
__device__ __forceinline__ void wait_async0() {
#if __has_builtin(__builtin_amdgcn_s_wait_asynccnt)
  __builtin_amdgcn_s_wait_asynccnt(0);
#else
  asm volatile("s_wait_asynccnt 0x0" ::: "memory");
#endif
}
#else
#define ASYNC_LDS 0
#endif

__device__ __forceinline__ v8f wmma4(v2f a, v2f b, v8f c) {
#if HAVE_WMMA_F32
  return __builtin_amdgcn_wmma_f32_16x16x4_f32(false, a, false, b, (short)0, c,
                                               false, false);
#else
  c[0] += a.x * b.x + a.y * b.y;  // placeholder for host pass only
  return c;
#endif
}

__device__ __forceinline__ float sigmoidf_(float x) {
  return 1.0f / (1.0f + expf(-x));
}

__device__ __forceinline__ float block_reduce_sum256(float v, float* red) {
  int tid = threadIdx.x;
  red[tid] = v;
  __syncthreads();
  for (int s = 128; s > 0; s >>= 1) {
    if (tid < s) red[tid] += red[tid + s];
    __syncthreads();
  }
  float r = red[0];
  __syncthreads();
  return r;
}

// packed-A element offset for logical (k, m):  (k>>1)*256 + m*2 + (k&1)
#define APACK(k, m) (((size_t)((k) >> 1) << 8) + ((m) << 1) + ((k) & 1))

// ---------------------------------------------------------------------------
// GEMM: D(128 x N) = A @ B,  A packed [K/2][128][2],  B row-major [k][ldb].
// Grid x = N/64 tiles, y = K-splits (deterministic partials).
// Block 256 thr = 8 waves; wave w owns M rows [16w,16w+16); 4 N-subtiles.
// Double-buffered LDS B tile (16K x 64N); async global->LDS when available.
// ---------------------------------------------------------------------------
#define GEMM_COMPUTE(CUR)                                        \
  {                                                              \
    const float* lbc = &lb[(CUR)][0];                            \
    _Pragma("unroll") for (int hk = 0; hk < 4; ++hk) {           \
      v2f af = *(const v2f*)(ap + hk * 512);                     \
      const int r0 = (4 * hk + 2 * khalf1) * 64 + (lane & 15);   \
      _Pragma("unroll") for (int nt = 0; nt < 4; ++nt) {         \
        v2f bf;                                                  \
        bf.x = lbc[r0 + 16 * nt];                                \
        bf.y = lbc[r0 + 64 + 16 * nt];                           \
        acc[nt] = wmma4(af, bf, acc[nt]);                        \
      }                                                          \
    }                                                            \
    ap += 2048;                                                  \
  }

__global__ __launch_bounds__(256) void gemm_kt_kernel(
    const float* __restrict__ Ap, const float* __restrict__ Bm, int ldb,
    int ncols, int k16_count, int k16_total, float* __restrict__ Dst, int ldd,
    long long dstride) {
  __shared__ float lb[2][16 * 64];
  const int tid = threadIdx.x;
  const int lane = tid & 31;
  const int w = tid >> 5;
  const int n0 = blockIdx.x * 64;
  const int k16_begin = blockIdx.y * k16_count;
  int k16_end = k16_begin + k16_count;
  if (k16_end > k16_total) k16_end = k16_total;
  float* Dp = Dst + (long long)blockIdx.y * dstride;

  // staging: one float4 per thread per tile (256 * 4 = 16x64)
  const int srow = tid >> 4;
  const int scol = (tid & 15) * 4;
  const bool svalid = (n0 + scol) < ncols;  // ncols % 4 == 0 -> all-or-none
  const float* sgp = Bm + (size_t)srow * ldb + n0 + scol;
  const int sdst = srow * 64 + scol;

  // zero-fill padded columns once (only the right-edge N tile)
  if (n0 + 64 > ncols) {
    for (int e = tid; e < 16 * 64; e += 256) {
      if (n0 + (e & 63) >= ncols) {
        lb[0][e] = 0.0f;
        lb[1][e] = 0.0f;
      }
    }
  }

  v8f acc[4] = {};
  const int mrow = w * 16 + (lane & 15);
  const int khalf1 = (lane >> 4);  // lanes 16-31: K base +2 (one k-pair)
  // A base: pair index 8*k16 + 2*hk + khalf1, element = pair*256 + mrow*2
  const float* ap =
      Ap + (((size_t)k16_begin * 8 + khalf1) << 8) + ((size_t)mrow << 1);

#if ASYNC_LDS
  if (svalid) async_stage16(sgp + (size_t)k16_begin * 16 * ldb, &lb[0][sdst]);
  wait_async0();
  __syncthreads();
  for (int k16 = k16_begin; k16 < k16_end; ++k16) {
    const int cur = k16 & 1;
    if ((k16 + 1 < k16_end) && svalid)
      async_stage16(sgp + (size_t)(k16 + 1) * 16 * ldb, &lb[cur ^ 1][sdst]);
    GEMM_COMPUTE(cur);
    wait_async0();
    __syncthreads();
  }
#else
  if (svalid) {
    float4 v0 = *(const float4*)(sgp + (size_t)k16_begin * 16 * ldb);
    *(float4*)&lb[0][sdst] = v0;
  }
  __syncthreads();
  for (int k16 = k16_begin; k16 < k16_end; ++k16) {
    const int cur = k16 & 1;
    float4 nv;
    const bool have_next = (k16 + 1 < k16_end) && svalid;
    if (have_next) nv = *(const float4*)(sgp + (size_t)(k16 + 1) * 16 * ldb);
    GEMM_COMPUTE(cur);
    if (have_next) *(float4*)&lb[cur ^ 1][sdst] = nv;
    __syncthreads();
  }
#endif

  // D: VGPR v -> M = v (+8 for lanes 16-31), N = lane%16
  const int mbase = w * 16 + 8 * (lane >> 4);
  const int nbase = n0 + (lane & 15);
#pragma unroll
  for (int nt = 0; nt < 4; ++nt) {
    int n = nbase + 16 * nt;
    if (n < ncols) {
#pragma unroll
      for (int v = 0; v < 8; ++v)
        Dp[(size_t)(mbase + v) * ldd + n] = acc[nt][v];
    }
  }
}

// ---------------------------------------------------------------------------
// GLU + LayerNorm of z = sum of KS_SYN GEMM partials (each 128 x 4096);
// writes state into trace ring slot t%25. trace layout: [b][d][32] padded.
// ---------------------------------------------------------------------------
__global__ __launch_bounds__(256) void glu_ln_kernel(
    const float* __restrict__ zp, const float* __restrict__ syn_b,
    const float* __restrict__ ln_g, const float* __restrict__ ln_b,
    float* __restrict__ trace, int t) {
  __shared__ float red[256];
  const int b = blockIdx.x, tid = threadIdx.x;
  const float* zr0 = zp + (size_t)b * (2 * DM);
  const float* zr1 = zr0 + (size_t)BB * (2 * DM);
  float g[8];
  float lsum = 0.0f;
#pragma unroll
  for (int j = 0; j < 8; ++j) {
    int d = tid + 256 * j;
    float av = zr0[d] + zr1[d] + syn_b[d];
    float bv = zr0[DM + d] + zr1[DM + d] + syn_b[DM + d];
    g[j] = av * sigmoidf_(bv);
    lsum += g[j];
  }
  float mu = block_reduce_sum256(lsum, red) * (1.0f / DM);
  float lv = 0.0f;
#pragma unroll
  for (int j = 0; j < 8; ++j) {
    float dlt = g[j] - mu;
    lv += dlt * dlt;
  }
  float var = block_reduce_sum256(lv, red) * (1.0f / DM);
  float inv = 1.0f / sqrtf(var + 1e-5f);
  const int slot = t % MEMN;
#pragma unroll
  for (int j = 0; j < 8; ++j) {
    int d = tid + 256 * j;
    float o = (g[j] - mu) * inv * ln_g[d] + ln_b[d];
    trace[((size_t)b * DM + d) * 32 + slot] = o;
  }
}

// ---------------------------------------------------------------------------
// trace_proc: per (b,d) tiny MLP 25 -> 64 (GLU->32) -> 2 (GLU->1).
// Block = 2 d x 128 b; per-d weights in LDS (broadcast reads across lanes).
// Writes act directly into the packed pre buffer (rows 512+d of A).
// ---------------------------------------------------------------------------
__global__ __launch_bounds__(256) void trace_proc_kernel(
    const float* __restrict__ trace, const float* __restrict__ w1t,
    const float* __restrict__ w2t, const float* __restrict__ tb1,
    const float* __restrict__ tb2, float* __restrict__ pre_p, int t) {
  __shared__ float lw1[2 * MEMN * 64];
  __shared__ float lw2[2 * 64];
  __shared__ float lb1[2 * 64];
  __shared__ float lb2[4];
  const int tid = threadIdx.x;
  const int d0 = blockIdx.x * 2;
  for (int i = tid; i < 2 * MEMN * 64; i += 256)
    lw1[i] = w1t[(size_t)d0 * (MEMN * 64) + i];
  for (int i = tid; i < 128; i += 256) lw2[i] = w2t[(size_t)d0 * 64 + i];
  for (int i = tid; i < 128; i += 256) lb1[i] = tb1[(size_t)d0 * 64 + i];
  if (tid < 4) lb2[tid] = tb2[d0 * 2 + tid];
  __syncthreads();

  const int dl = tid >> 7;
  const int b = tid & 127;
  const int d = d0 + dl;
  const float* w1 = lw1 + dl * (MEMN * 64);
  const float* w2 = lw2 + dl * 64;
  const float* b1 = lb1 + dl * 64;

  float tr[MEMN];
  const float* trow = trace + ((size_t)b * DM + d) * 32;
  const int base = (t + 1) % MEMN;
#pragma unroll
  for (int m = 0; m < MEMN; ++m) {
    int sl = base + m;
    sl -= (sl >= MEMN) ? MEMN : 0;
    tr[m] = trow[sl];
  }

  float o0 = lb2[dl * 2 + 0], o1 = lb2[dl * 2 + 1];
  for (int h = 0; h < HIDN; ++h) {
    float sa = b1[h], sb = b1[h + HIDN];
#pragma unroll
    for (int m = 0; m < MEMN; ++m) {
      sa = fmaf(tr[m], w1[m * 64 + h], sa);
      sb = fmaf(tr[m], w1[m * 64 + h + HIDN], sb);
    }
    float gg = sa * sigmoidf_(sb);
    o0 = fmaf(gg, w2[h * 2 + 0], o0);
    o1 = fmaf(gg, w2[h * 2 + 1], o1);
  }
  // act row of packed pre: logical k = 512 + d
  pre_p[APACK(512 + d, b)] = o0 * sigmoidf_(o1);
}

// ---------------------------------------------------------------------------
// pairwise EMA + sync (packed [s/2][128][2] for the out-GEMM A side).
// Block = 2 s x 128 b.
// ---------------------------------------------------------------------------
__global__ __launch_bounds__(256) void pairwise_kernel(
    const float* __restrict__ pre_p, const int* __restrict__ gl,
    const int* __restrict__ gr, const float* __restrict__ r,
    float* __restrict__ alpha, float* __restrict__ beta,
    float* __restrict__ sync_p) {
  const int tid = threadIdx.x;
  const int sl = tid >> 7;
  const int s = blockIdx.x * 2 + sl;
  const int b = tid & 127;
  const int dl = gl[s], dr = gr[s];
  const float av = pre_p[APACK(512 + dl, b)];
  const float bv = pre_p[APACK(512 + dr, b)];
  const float rv = r[s];
  const size_t idx = (size_t)s * BB + b;
  const float al = fmaf(rv, alpha[idx], av * bv);
  const float be = fmaf(rv, beta[idx], 1.0f);
  alpha[idx] = al;
  beta[idx] = be;
  sync_p[(size_t)blockIdx.x * 256 + 2 * b + sl] = al / sqrtf(be);
}

// ---------------------------------------------------------------------------
// K-split reduction + bias + log-softmax entropy + strided output writes
// (preds [b][o][t], cert [b][2][t]).  One block per batch row.
// ---------------------------------------------------------------------------
__global__ __launch_bounds__(256) void softmax_out_kernel(
    const float* __restrict__ part, const float* __restrict__ out_b,
    float* __restrict__ dout, int t) {
  __shared__ float red[256];
  __shared__ float sp[OUTD];
  const int b = blockIdx.x, tid = threadIdx.x;
  float lmax = -3.4e38f;
  for (int o = tid; o < OUTD; o += 256) {
    float v = out_b[o];
#pragma unroll
    for (int ks = 0; ks < KS_OUT; ++ks)
      v += part[((size_t)ks * BB + b) * 1024 + o];
    sp[o] = v;
    lmax = fmaxf(lmax, v);
  }
  red[tid] = lmax;
  __syncthreads();
  for (int s = 128; s > 0; s >>= 1) {
    if (tid < s) red[tid] = fmaxf(red[tid], red[tid + s]);
    __syncthreads();
  }
  const float mx = red[0];
  __syncthreads();
  float lsum = 0.0f;
  for (int o = tid; o < OUTD; o += 256) lsum += expf(sp[o] - mx);
  const float logZ = mx + logf(block_reduce_sum256(lsum, red));
  float lne = 0.0f;
  for (int o = tid; o < OUTD; o += 256) {
    float lp = sp[o] - logZ;
    lne += expf(lp) * lp;
  }
  const float nes = block_reduce_sum256(lne, red);
  const float ne = -nes / 6.907755278982137f;  // log(1000)
  float* po = dout + (size_t)b * OUTD * ITERS;
  for (int o = tid; o < OUTD; o += 256) po[(size_t)o * ITERS + t] = sp[o];
  if (tid == 0) {
    float* pc = dout + (size_t)BB * OUTD * ITERS + (size_t)b * 2 * ITERS;
    pc[t] = ne;
    pc[ITERS + t] = 1.0f - ne;
  }
}

// ------------------------------- init kernels ------------------------------
__global__ void k_r_kernel(const float* __restrict__ decay,
                           float* __restrict__ r) {
  int i = blockIdx.x * 256 + threadIdx.x;
  if (i < SYNCN) {
    float dv = fminf(fmaxf(decay[i], 0.0f), 15.0f);
    r[i] = expf(-dv);
  }
}

__global__ void k_pairs_kernel(const int* __restrict__ idx_left,
                               const int* __restrict__ idx_right,
                               int* __restrict__ gl, int* __restrict__ gr) {
  int gid = blockIdx.x * 256 + threadIdx.x;
  if (gid >= NSY * NSY) return;
  int i = gid >> 8, j = gid & 255;
  if (j < i) return;
  int s = i * NSY - (i * (i - 1)) / 2 + (j - i);  // triu row-major index
  gl[s] = idx_left[i];
  gr[s] = idx_right[j];
}

__global__ void k_prex_kernel(const float* __restrict__ x,
                              float* __restrict__ pre_p) {
  int gid = blockIdx.x * 256 + threadIdx.x;  // k*128 + b, k < 512
  if (gid >= DIN * BB) return;
  int k = gid >> 7, b = gid & 127;
  pre_p[APACK(k, b)] = x[(size_t)b * DIN + k];
}

__global__ void k_w1t_kernel(const float* __restrict__ w1,
                             float* __restrict__ w1t) {
  int gid = blockIdx.x * 256 + threadIdx.x;
  if (gid >= DM * MEMN * 64) return;
  int d = gid / (MEMN * 64);
  int rem = gid % (MEMN * 64);
  int m = rem >> 6, h = rem & 63;
  w1t[gid] = w1[((size_t)m * 64 + h) * DM + d];
}

__global__ void k_w2t_kernel(const float* __restrict__ w2,
                             float* __restrict__ w2t) {
  int gid = blockIdx.x * 256 + threadIdx.x;
  if (gid >= DM * 64) return;
  int d = gid >> 6;
  int rem = gid & 63;
  int h = rem >> 1, o = rem & 1;
  w2t[gid] = w2[((size_t)h * 2 + o) * DM + d];
}

__global__ void k_trace_init_kernel(const float* __restrict__ st,
                                    float* __restrict__ trace) {
  int gid = blockIdx.x * 256 + threadIdx.x;  // b*DM + d
  if (gid >= BB * DM) return;
  int d = gid & (DM - 1);
  float* trow = trace + (size_t)gid * 32;
#pragma unroll
  for (int m = 0; m < MEMN; ++m) trow[m] = st[(size_t)d * MEMN + m];
}

__global__ void k_act_init_kernel(const float* __restrict__ sa,
                                  float* __restrict__ pre_p) {
  int gid = blockIdx.x * 256 + threadIdx.x;  // d*128 + b
  if (gid >= DM * BB) return;
  int d = gid >> 7, b = gid & 127;
  pre_p[APACK(512 + d, b)] = sa[d];
}

__global__ void k_alpha_init_kernel(const float* __restrict__ sa,
                                    const int* __restrict__ gl,
                                    const int* __restrict__ gr,
                                    float* __restrict__ alpha,
                                    float* __restrict__ beta) {
  int gid = blockIdx.x * 256 + threadIdx.x;  // s*128 + b
  if (gid >= SYNCN * BB) return;
  int s = gid >> 7;
  alpha[gid] = sa[gl[s]] * sa[gr[s]];
  beta[gid] = 1.0f;
}

// ---------------------------------------------------------------------------
extern "C" void kernel_launch(void* const* d_in, const int* in_sizes, int n_in,
                              void* d_out, int out_size, void* d_ws,
                              size_t ws_size, hipStream_t stream) {
  (void)in_sizes; (void)n_in; (void)out_size; (void)ws_size;

  const float* x      = (const float*)d_in[0];
  const float* syn_w  = (const float*)d_in[1];
  const float* syn_b  = (const float*)d_in[2];
  const float* ln_g   = (const float*)d_in[3];
  const float* ln_b   = (const float*)d_in[4];
  const float* tp_w1  = (const float*)d_in[5];
  const float* tp_b1  = (const float*)d_in[6];
  const float* tp_w2  = (const float*)d_in[7];
  const float* tp_b2  = (const float*)d_in[8];
  const float* sa     = (const float*)d_in[9];
  const float* st     = (const float*)d_in[10];
  const float* decay  = (const float*)d_in[11];
  const float* out_w  = (const float*)d_in[12];
  const float* out_b  = (const float*)d_in[13];
  const int* idx_left  = (const int*)d_in[14];
  const int* idx_right = (const int*)d_in[15];
  // d_in[16] = iterations (fixed at 24 by the reference)

  float* ws = (float*)d_ws;
  size_t off = 0;
  float* trace = ws + off;  off += (size_t)BB * DM * 32;        // ring, padded
  float* pre_p = ws + off;  off += (size_t)(DIN + DM) * BB;     // packed A syn
  float* zp    = ws + off;  off += (size_t)KS_SYN * BB * 2 * DM;  // syn parts
  float* alpha = ws + off;  off += (size_t)SYNCN * BB;          // [s][b]
  float* beta  = ws + off;  off += (size_t)SYNCN * BB;
  float* syncp = ws + off;  off += (size_t)SYNCN * BB;          // packed A out
  float* rbuf  = ws + off;  off += SYNCN;
  int*   gl    = (int*)(ws + off); off += SYNCN;
  int*   gr    = (int*)(ws + off); off += SYNCN;
  float* w1t   = ws + off;  off += (size_t)DM * MEMN * 64;      // [d][m][h]
  float* w2t   = ws + off;  off += (size_t)DM * 64;             // [d][h][o]
  float* part  = ws + off;  off += (size_t)KS_OUT * BB * 1024;  // out parts
  // total ~28.0M floats (~112 MB)

  // ---- init (re-run every call: deterministic, no persistent state) ----
  k_r_kernel<<<(SYNCN + 255) / 256, 256, 0, stream>>>(decay, rbuf);
  k_pairs_kernel<<<(NSY * NSY) / 256, 256, 0, stream>>>(idx_left, idx_right,
                                                        gl, gr);
  k_prex_kernel<<<(DIN * BB) / 256, 256, 0, stream>>>(x, pre_p);
  k_w1t_kernel<<<(DM * MEMN * 64) / 256, 256, 0, stream>>>(tp_w1, w1t);
  k_w2t_kernel<<<(DM * 64) / 256, 256, 0, stream>>>(tp_w2, w2t);
  k_trace_init_kernel<<<(BB * DM) / 256, 256, 0, stream>>>(st, trace);
  k_act_init_kernel<<<(DM * BB) / 256, 256, 0, stream>>>(sa, pre_p);
  k_alpha_init_kernel<<<(SYNCN * BB) / 256, 256, 0, stream>>>(sa, gl, gr,
                                                              alpha, beta);

  // ---- 24 recurrent steps ----
  for (int t = 0; t < ITERS; ++t) {
    // z = [x, act] @ syn_w   (M=128, N=4096, K=2560; 2 K-splits x 64 N)
    gemm_kt_kernel<<<dim3(64, KS_SYN), 256, 0, stream>>>(
        pre_p, syn_w, 2 * DM, 2 * DM, /*k16_count=*/160 / KS_SYN,
        /*k16_total=*/160, zp, 2 * DM, (long long)BB * 2 * DM);
    glu_ln_kernel<<<BB, 256, 0, stream>>>(zp, syn_b, ln_g, ln_b, trace, t);
    trace_proc_kernel<<<DM / 2, 256, 0, stream>>>(trace, w1t, w2t, tp_b1,
                                                  tp_b2, pre_p, t);
    pairwise_kernel<<<SYNCN / 2, 256, 0, stream>>>(pre_p, gl, gr, rbuf, alpha,
                                                   beta, syncp);
    // pred_part = sync @ out_w  (M=128, N=1000, K=32896; 16 K-splits x 16 N)
    gemm_kt_kernel<<<dim3(16, KS_OUT), 256, 0, stream>>>(
        syncp, out_w, OUTD, OUTD, /*k16_count=*/129, /*k16_total=*/2056, part,
        1024, (long long)BB * 1024);
    softmax_out_kernel<<<BB, 256, 0, stream>>>(part, out_b, (float*)d_out, t);
  }
}